// Quanvolution_73263552135204
// MI455X (gfx1250) — compile-verified
//
#include <hip/hip_runtime.h>
#include <math.h>

typedef __attribute__((ext_vector_type(2))) float v2f;
typedef __attribute__((ext_vector_type(8))) float v8f;

#define HALF_PI 1.57079632679489662f

// ---------------------------------------------------------------------------
// Prologue: build the 16x16 real unitary U = RING * kron(RY(th0)..RY(th3))
// from rand_params (4 floats). One entry per thread (256 threads).
// U is stored row-major: U[i*16 + j].
// ---------------------------------------------------------------------------
__global__ void build_unitary_kernel(const float* __restrict__ rp,
                                     float* __restrict__ U) {
    int t = threadIdx.x;       // 0..255
    int m = t >> 4;            // Kkron row (pre-ring basis index)
    int j = t & 15;            // column
    float val = 1.0f;
#pragma unroll
    for (int w = 0; w < 4; ++w) {
        float th = 0.5f * rp[w];
        float c = __cosf(th), s = __sinf(th);
        int bm = (m >> (3 - w)) & 1;
        int bj = (j >> (3 - w)) & 1;
        // RY = [[c,-s],[s,c]]  -> entry [bm][bj]
        float e = (bm == bj) ? c : (bm ? s : -s);
        val *= e;
    }
    // CNOT ring permutation (wire0 = MSB): CNOT(0,1),(1,2),(2,3),(3,0)
    int b0 = (m >> 3) & 1, b1 = (m >> 2) & 1, b2 = (m >> 1) & 1, b3 = m & 1;
    b1 ^= b0; b2 ^= b1; b3 ^= b2; b0 ^= b3;
    int i = (b0 << 3) | (b1 << 2) | (b2 << 1) | b3;
    U[i * 16 + j] = val;
}

// ---------------------------------------------------------------------------
// Main kernel: each wave32 processes 16 patches.
//   D(16x16) = U(16x16) x S(16x16patches) via 4 chained V_WMMA_F32_16X16X4_F32
//   A = U (loop-invariant), B = per-patch product states (one patch per lane).
//   D layout gives each lane 8 y-values of its own patch -> in-register signed
//   reduction + one shfl_xor(16) merge, then a single float4 store per patch.
// ---------------------------------------------------------------------------
__global__ void __launch_bounds__(256)
quanv_wmma_kernel(const float* __restrict__ x, const float* __restrict__ U,
                  float* __restrict__ out, int B) {
    const int lane  = (int)(threadIdx.x & 31);
    const int halfw = lane >> 4;    // 0: K%4 in {0,1} (b2=0) | 1: {2,3} (b2=1)
    const int l     = lane & 15;    // patch-in-group (B cols) / U row (A rows)

    const unsigned NP      = (unsigned)B * 196u;       // total patches
    const unsigned ngroups = (NP + 15u) >> 4;
    unsigned wave   = (unsigned)(blockIdx.x * blockDim.x + threadIdx.x) >> 5;
    const unsigned nwaves = (unsigned)(gridDim.x * blockDim.x) >> 5;

    // Loop-invariant A fragments: A[M=i][K=j] = U[i][j].
    // f32 16x4 A layout: lane(M=l) VGPR0/1 = K base+0/1 (low), base+2/3 (high).
    v2f ufrag[4];
#pragma unroll
    for (int c = 0; c < 4; ++c) {
        int base = 4 * c + 2 * halfw;
        ufrag[c].x = U[l * 16 + base + 0];
        ufrag[c].y = U[l * 16 + base + 1];
    }

    for (unsigned g = wave; g < ngroups; g += nwaves) {
        unsigned p = g * 16u + (unsigned)l;     // patch for this lane pair
        if (p >= NP) p = NP - 1u;               // clamp (keep EXEC full for WMMA)
        unsigned b  = p / 196u;
        unsigned r  = p - b * 196u;
        unsigned jj = r / 14u;
        unsigned kc = r - jj * 14u;
        const float* px = x + (b * 784u + jj * 56u + kc * 2u);
        float2 row0 = *(const float2*)(px);        // pixels (j,k),(j,k+1)
        float2 row1 = *(const float2*)(px + 28);   // pixels (j+1,k),(j+1,k+1)

        float c0, s0, c1, s1, c2, s2, c3, s3;
        __sincosf(HALF_PI * row0.x, &s0, &c0);
        __sincosf(HALF_PI * row0.y, &s1, &c1);
        __sincosf(HALF_PI * row1.x, &s2, &c2);
        __sincosf(HALF_PI * row1.y, &s3, &c3);

        // B[K=j][N=patch]: j = b0*8 + b1*4 + b2*2 + b3; chunk c = (b0,b1),
        // lane half = b2, vector element = b3.
        float t00 = c0 * c1, t01 = c0 * s1, t10 = s0 * c1, t11 = s0 * s1;
        float w2 = halfw ? s2 : c2;
        float e0 = t00 * w2, e1 = t01 * w2, e2 = t10 * w2, e3 = t11 * w2;
        v2f s0v, s1v, s2v, s3v;
        s0v.x = e0 * c3; s0v.y = e0 * s3;
        s1v.x = e1 * c3; s1v.y = e1 * s3;
        s2v.x = e2 * c3; s2v.y = e2 * s3;
        s3v.x = e3 * c3; s3v.y = e3 * s3;

        v8f acc = {0.f, 0.f, 0.f, 0.f, 0.f, 0.f, 0.f, 0.f};
        acc = __builtin_amdgcn_wmma_f32_16x16x4_f32(false, ufrag[0], false, s0v,
                                                    (short)0, acc, false, false);
        acc = __builtin_amdgcn_wmma_f32_16x16x4_f32(false, ufrag[1], false, s1v,
                                                    (short)0, acc, false, false);
        acc = __builtin_amdgcn_wmma_f32_16x16x4_f32(false, ufrag[2], false, s2v,
                                                    (short)0, acc, false, false);
        acc = __builtin_amdgcn_wmma_f32_16x16x4_f32(false, ufrag[3], false, s3v,
                                                    (short)0, acc, false, false);

        // Lane n (<16) holds y[i=0..7] of patch g*16+n in acc[0..7];
        // lane n+16 holds y[i=8..15] of the same patch.
        float q[8];
#pragma unroll
        for (int rr = 0; rr < 8; ++rr) { float y = acc[rr]; q[rr] = y * y; }

        // In-register partial Hadamard over dim bits b1,b2,b3 (i = r + 8*half):
        float u0 = q[0] + q[1], u1 = q[2] + q[3], u2 = q[4] + q[5], u3 = q[6] + q[7];
        float d0 = q[0] - q[1], d1 = q[2] - q[3], d2 = q[4] - q[5], d3 = q[6] - q[7];
        float w3p = (d0 + d1) + (d2 + d3);          // sign on b3 (wire 3)
        float w2p = (u0 - u1) + (u2 - u3);          // sign on b2 (wire 2)
        float av  = u0 + u1, bv = u2 + u3;
        float w1p = av - bv;                        // sign on b1 (wire 1)
        float tot = av + bv;                        // plain half-sum

        // Merge halves (b0 = half): one shuffle per output.
        float o3 = w3p + __shfl_xor(w3p, 16, 32);
        float o2 = w2p + __shfl_xor(w2p, 16, 32);
        float o1 = w1p + __shfl_xor(w1p, 16, 32);
        float o0 = tot - __shfl_xor(tot, 16, 32);   // valid in lanes < 16

        if (halfw == 0) {
            unsigned patch = g * 16u + (unsigned)l;
            if (patch < NP) {
                float4 res = make_float4(o0, o1, o2, o3);
                *(float4*)(out + (size_t)patch * 4u) = res;
            }
        }
    }
}

// ---------------------------------------------------------------------------
extern "C" void kernel_launch(void* const* d_in, const int* in_sizes, int n_in,
                              void* d_out, int out_size, void* d_ws, size_t ws_size,
                              hipStream_t stream) {
    const float* x  = (const float*)d_in[0];   // [B,28,28,1] f32
    const float* rp = (const float*)d_in[1];   // [1,4] f32
    float* out = (float*)d_out;                // [B,14,14,4] f32
    float* U   = (float*)d_ws;                 // 256 floats

    int B = in_sizes[0] / 784;

    build_unitary_kernel<<<1, 256, 0, stream>>>(rp, U);

    long NP = (long)B * 196;
    long ngroups = (NP + 15) >> 4;             // 16 patches per wave
    long wavesPerBlock = 8;                    // 256 threads
    long blocksNeeded = (ngroups + wavesPerBlock - 1) / wavesPerBlock;
    int blocks = (int)(blocksNeeded < 8192 ? blocksNeeded : 8192);
    if (blocks < 1) blocks = 1;
    quanv_wmma_kernel<<<blocks, 256, 0, stream>>>(x, U, out, B);
}